// MonteCarloPooling_52913997087065
// MI455X (gfx1250) — compile-verified
//
#include <hip/hip_runtime.h>
#include <hip/hip_bf16.h>
#include <stdint.h>

// MonteCarloPooling: out[b,c,hp,wp] = inputs[b=c, r>>2, 2*hp+((r>>1)&1), 2*wp+(r&1)],
// r = rand_idx[hp,wp,c]; broadcast over b. B=C=64, H=W=128, Hp=Wp=64.
//
// Memory-bound: 64 MiB output store dominates (~3-4.5 us at 23.3 TB/s).
// CDNA5 path: stage 4 KiB picked-tile in LDS, then broadcast to the 64 batch
// copies with the Tensor Data Mover (iterate mode, lds_addr_increment=0).

typedef unsigned int u32x4 __attribute__((ext_vector_type(4)));
typedef int          i32x4 __attribute__((ext_vector_type(4)));
typedef int          i32x8 __attribute__((ext_vector_type(8)));

#if __has_builtin(__builtin_amdgcn_tensor_store_from_lds)
#define MC_HAVE_TDM 1
#else
#define MC_HAVE_TDM 0
#endif

__global__ __launch_bounds__(256) void mc_pool_kernel(
    const float* __restrict__ in,   // [64,64,128,128]
    const int*   __restrict__ idx,  // [64,64,64] (hp,wp,c)
    float*       __restrict__ out)  // [64,64,64,64] (b,c,hp,wp)
{
    __shared__ float tile[1024];          // 4 KiB picked tile for this block

    const int tid    = threadIdx.x;       // 0..255
    const int blk    = blockIdx.x;        // 0..255
    const int c      = blk >> 2;          // output channel (== source batch row)
    const int hpBase = (blk & 3) << 4;    // 16 hp rows per block
    const int e0     = tid << 2;          // first of 4 elements for this thread

    // ---- Phase 1: gather 4 picked values, stage in LDS ----
    float4 v;
    float* vp = reinterpret_cast<float*>(&v);
#pragma unroll
    for (int j = 0; j < 4; ++j) {
        const int e  = e0 + j;
        const int hp = hpBase + (e >> 6);
        const int wp = e & 63;
        const int r  = idx[(hp * 64 + wp) * 64 + c];   // in [0, 256)
        const int cs = r >> 2;
        const int ph = (r >> 1) & 1;
        const int pw = r & 1;
        vp[j] = in[(((c * 64 + cs) * 128) + (hp * 2 + ph)) * 128 + (wp * 2 + pw)];
    }
    reinterpret_cast<float4*>(tile)[tid] = v;
    __syncthreads();

#if MC_HAVE_TDM
    // ---- Phase 2: TDM broadcast. Each of 8 waves stores the same 4 KiB LDS
    // tile to 8 batch copies via one tensor_store_from_lds (iterate 8x,
    // lds_addr_increment = 0, global_addr_increment = 1 MiB batch stride).
    const int wave = __builtin_amdgcn_readfirstlane((int)(tid >> 5));   // 0..7
    const unsigned lds_off = (unsigned)(uintptr_t)(void*)tile;          // LDS byte offset
    const unsigned long long ga =
        (unsigned long long)(uintptr_t)out
        + (unsigned long long)blk * 4096ull               // tile offset within one batch image
        + (unsigned long long)wave * (8ull * 1048576ull); // this wave's first batch copy

    // D# group 0 (128b): count=1 | lds_addr | global_addr[56:0] | type=2
    u32x4 g0;
    g0[0] = 1u;
    g0[1] = lds_off;
    g0[2] = (unsigned)(ga & 0xffffffffull);
    g0[3] = ((unsigned)(ga >> 32) & 0x01ffffffu) | (2u << 30);

    // D# group 1 (256b)
    i32x8 g1;
    g1[0] = (2 << 16) | (1 << 19);   // workgroup_mask=0, data_size=4B, iterate_enable=1
    g1[1] = (int)(1024u << 16);      // atomic_barrier_addr=0 ; tensor_dim0 = 1024 (lo16)
    g1[2] = (int)(1u << 16);         // tensor_dim0 hi=0 ; tensor_dim1 = 1
    g1[3] = (int)(1024u << 16);      // tensor_dim1 hi=0 ; tile_dim0 = 1024
    g1[4] = 1;                       // tile_dim1 = 1 ; tile_dim2 = 0
    g1[5] = 1024;                    // tensor_dim0_stride = 1024
    g1[6] = 0;
    g1[7] = 0;

    // D# group 2 (128b), iterate mode
    i32x4 g2;
    g2[0] = 1;                       // tensor_dim2 (unused)
    g2[1] = 0;                       // lds_addr_increment = 0  -> re-read same tile
    g2[2] = 262144;                  // global_addr_increment = 262144 elems = 1 MiB
    g2[3] = (7 << 16);               // gai[47:32]=0 ; iterate_count = 7 -> 8 iterations

    i32x4 g3 = {0, 0, 0, 0};         // group 3: dims 3/4 unused

    // 6-arg clang-23 form: (u32x4, i32x8, i32x4, i32x4, i32x8, i32 cpol).
    // The trailing i32x8 is uncharacterized; zero-filled (NULL/no-op encoding).
    i32x8 g4 = {0, 0, 0, 0, 0, 0, 0, 0};
    __builtin_amdgcn_tensor_store_from_lds(g0, g1, g2, g3, g4, 0);
#if __has_builtin(__builtin_amdgcn_s_wait_tensorcnt)
    __builtin_amdgcn_s_wait_tensorcnt(0);
#endif
#else
    // ---- Fallback: plain vectorized broadcast stores ----
    const float4 val = reinterpret_cast<const float4*>(tile)[tid];
    float* dst = out + (size_t)blk * 1024 + (size_t)e0;
#pragma unroll 4
    for (int b = 0; b < 64; ++b) {
        *reinterpret_cast<float4*>(dst + (size_t)b * 262144) = val;
    }
#endif
}

extern "C" void kernel_launch(void* const* d_in, const int* in_sizes, int n_in,
                              void* d_out, int out_size, void* d_ws, size_t ws_size,
                              hipStream_t stream) {
    (void)in_sizes; (void)n_in; (void)d_ws; (void)ws_size; (void)out_size;
    const float* in  = (const float*)d_in[0];   // [64,64,128,128] f32
    const int*   idx = (const int*)d_in[1];     // [64,64,64] i32
    float*       out = (float*)d_out;           // [64,64,64,64] f32
    mc_pool_kernel<<<256, 256, 0, stream>>>(in, idx, out);
}